// LinearAttention_49349174231208
// MI455X (gfx1250) — compile-verified
//
#include <hip/hip_runtime.h>

// ---------------------------------------------------------------------------
// LinearAttention step on MI455X (gfx1250), wave32.
//
// Roofline: Si state stream (537 MB in + 537 MB out) dominates => ~46 us at
// 23.3 TB/s. All GEMMs together are ~1.6 GFLOP => noise. Strategy:
//   A) WMMA f32 16x16x4 projections (q,k,v share the A fragment)
//   B) single fused pass over Si: rank-1 update + V readout, float4 traffic
//   C) WMMA f32 output projection
// ---------------------------------------------------------------------------

typedef __attribute__((ext_vector_type(2))) float v2f;
typedef __attribute__((ext_vector_type(8))) float v8f;

#define NTOK 512
#define NH   4
#define RTOT (NTOK*NH)   // 2048 independent (n,h) rows
#define EDIM 256
#define EPS_ 1e-6f

// d_out float offsets: (out, Si_new, Zi_new) concatenated in return order
#define OFF_SI  ((size_t)RTOT*EDIM)                       // 524288
#define OFF_ZI  (OFF_SI + (size_t)RTOT*EDIM*EDIM)         // 134742016

// ---------------------------------------------------------------------------
// Kernel A: q/k/v projections with V_WMMA_F32_16X16X4_F32.
// One wave computes one 16x16 tile of all three outputs (A fragment reused).
// A lane layout (ISA 7.12.2, 32-bit A 16x4): row = lane%16, VGPR0 holds
// K = 2*(lane/16), VGPR1 holds K+1  -> a contiguous float2 from the x row.
// B = W^T, so B per lane is a contiguous float2 from row (tileN*16 + lane%16)
// of W. C/D: lane L, vgpr j -> row j + 8*(L/16), col L%16.
// ---------------------------------------------------------------------------
__global__ __launch_bounds__(256)
void la_proj_kernel(const float* __restrict__ x,
                    const float* __restrict__ Zi,
                    const float* __restrict__ Wq, const float* __restrict__ bq,
                    const float* __restrict__ Wk, const float* __restrict__ bk,
                    const float* __restrict__ Wv, const float* __restrict__ bv,
                    float* __restrict__ kq_out, float* __restrict__ kk_out,
                    float* __restrict__ v_out,  float* __restrict__ zi_new_out)
{
    const int lane  = threadIdx.x & 31;
    const int wid   = blockIdx.x * (blockDim.x >> 5) + (threadIdx.x >> 5);
    const int tileM = wid >> 4;    // 0..127  (rows of 16 over RTOT=2048)
    const int tileN = wid & 15;    // 0..15   (cols of 16 over E=256)
    const int half  = lane >> 4;   // 0|1
    const int l16   = lane & 15;

    const int rowA = tileM * 16 + l16;
    const int colB = tileN * 16 + l16;

    const float* xr  = x  + (size_t)rowA * EDIM;
    const float* wqr = Wq + (size_t)colB * EDIM;
    const float* wkr = Wk + (size_t)colB * EDIM;
    const float* wvr = Wv + (size_t)colB * EDIM;

    v8f accq = {}; v8f acck = {}; v8f accv = {};
    #pragma unroll 4
    for (int it = 0; it < EDIM / 4; ++it) {
        const int k0 = it * 4 + half * 2;
        v2f a   = *(const v2f*)(xr  + k0);
        v2f b_q = *(const v2f*)(wqr + k0);
        v2f b_k = *(const v2f*)(wkr + k0);
        v2f b_v = *(const v2f*)(wvr + k0);
        accq = __builtin_amdgcn_wmma_f32_16x16x4_f32(false, a, false, b_q,
                                                     (short)0, accq, false, false);
        acck = __builtin_amdgcn_wmma_f32_16x16x4_f32(false, a, false, b_k,
                                                     (short)0, acck, false, false);
        accv = __builtin_amdgcn_wmma_f32_16x16x4_f32(false, a, false, b_v,
                                                     (short)0, accv, false, false);
    }

    const float bqc = bq[colB], bkc = bk[colB], bvc = bv[colB];
    #pragma unroll
    for (int j = 0; j < 8; ++j) {
        const int row = tileM * 16 + j + half * 8;
        const size_t idx = (size_t)row * EDIM + colB;
        const float qv = accq[j] + bqc;
        const float kv = acck[j] + bkc;
        const float vv = accv[j] + bvc;
        // elu(t)+1 = t+1 (t>0) else exp(t)
        const float kqv = (qv > 0.f) ? (qv + 1.f) : __expf(qv);
        const float kkv = (kv > 0.f) ? (kv + 1.f) : __expf(kv);
        kq_out[idx]     = kqv;
        kk_out[idx]     = kkv;
        v_out[idx]      = vv;
        zi_new_out[idx] = Zi[idx] + kkv;   // Zi_new
    }
}

// ---------------------------------------------------------------------------
// Kernel B: the bandwidth kernel. One block per (n,h) row r.
//   Zinv      = 1/(kq . Zi_new + eps)           (LDS tree reduction)
//   Si_new    = Si + kk (outer) v                (streamed, float4)
//   V[m]      = Zinv * sum_d kq[d]*Si_new[d][m]  (fused into the same stream)
// Thread t: row-group rg = t>>6 (4 rows in flight), columns 4*(t&63)..+3.
// Per loop iteration the block touches one contiguous 4 KB span of Si.
// ---------------------------------------------------------------------------
__global__ __launch_bounds__(256)
void la_state_kernel(const float* __restrict__ Si,
                     const float* __restrict__ kq_in,
                     const float* __restrict__ kk_in,
                     const float* __restrict__ v_in,
                     const float* __restrict__ zi_new,
                     float* __restrict__ Si_new,
                     float* __restrict__ V_out)
{
    __shared__ __attribute__((aligned(16))) float s_kk[EDIM];
    __shared__ __attribute__((aligned(16))) float s_kq[EDIM];
    __shared__ __attribute__((aligned(16))) float s_v [EDIM];
    __shared__ __attribute__((aligned(16))) float s_red[4 * EDIM];
    __shared__ float s_z[256];

    const int r = blockIdx.x;
    const int t = threadIdx.x;
    const size_t rbase = (size_t)r * EDIM;

    const float kqt = kq_in[rbase + t];
    s_kq[t] = kqt;
    s_kk[t] = kk_in[rbase + t];
    s_v [t] = v_in [rbase + t];
    s_z [t] = kqt * zi_new[rbase + t];
    __syncthreads();
    #pragma unroll
    for (int s = 128; s > 0; s >>= 1) {
        if (t < s) s_z[t] += s_z[t + s];
        __syncthreads();
    }
    const float Zinv = 1.0f / (s_z[0] + EPS_);

    const int rg = t >> 6;            // 0..3 : which of 4 concurrent rows
    const int c4 = (t & 63) << 2;     // column base 0,4,...,252
    const float4 v4 = *(const float4*)&s_v[c4];
    float4 vacc = make_float4(0.f, 0.f, 0.f, 0.f);

    const float* sin  = Si     + (size_t)r * EDIM * EDIM;
    float*       sout = Si_new + (size_t)r * EDIM * EDIM;

    #pragma unroll 4
    for (int dd = 0; dd < EDIM / 4; ++dd) {
        const int d = dd * 4 + rg;
        const float kkd = s_kk[d];
        const float kqd = s_kq[d];
        const size_t off = (size_t)d * EDIM + c4;
        float4 si = *(const float4*)(sin + off);    // global_load_b128
        float4 sn;
        sn.x = si.x + kkd * v4.x;
        sn.y = si.y + kkd * v4.y;
        sn.z = si.z + kkd * v4.z;
        sn.w = si.w + kkd * v4.w;
        *(float4*)(sout + off) = sn;                // global_store_b128
        vacc.x += kqd * sn.x;
        vacc.y += kqd * sn.y;
        vacc.z += kqd * sn.z;
        vacc.w += kqd * sn.w;
    }

    *(float4*)&s_red[rg * EDIM + c4] = vacc;
    __syncthreads();
    if (t < 64) {
        #pragma unroll
        for (int i = 0; i < 4; ++i) {
            const int col = t * 4 + i;
            float Vv = s_red[col] + s_red[EDIM + col] +
                       s_red[2 * EDIM + col] + s_red[3 * EDIM + col];
            V_out[rbase + col] = Vv * Zinv;
        }
    }
}

// ---------------------------------------------------------------------------
// Kernel C: out = V * Wo^T + bo   (same WMMA tiling as kernel A)
// ---------------------------------------------------------------------------
__global__ __launch_bounds__(256)
void la_out_kernel(const float* __restrict__ Vin,
                   const float* __restrict__ Wo, const float* __restrict__ bo,
                   float* __restrict__ out)
{
    const int lane  = threadIdx.x & 31;
    const int wid   = blockIdx.x * (blockDim.x >> 5) + (threadIdx.x >> 5);
    const int tileM = wid >> 4;
    const int tileN = wid & 15;
    const int half  = lane >> 4;
    const int l16   = lane & 15;

    const int rowA = tileM * 16 + l16;
    const int colB = tileN * 16 + l16;

    const float* vr  = Vin + (size_t)rowA * EDIM;
    const float* wor = Wo  + (size_t)colB * EDIM;

    v8f acc = {};
    #pragma unroll 4
    for (int it = 0; it < EDIM / 4; ++it) {
        const int k0 = it * 4 + half * 2;
        v2f a = *(const v2f*)(vr  + k0);
        v2f b = *(const v2f*)(wor + k0);
        acc = __builtin_amdgcn_wmma_f32_16x16x4_f32(false, a, false, b,
                                                    (short)0, acc, false, false);
    }

    const float boc = bo[colB];
    #pragma unroll
    for (int j = 0; j < 8; ++j) {
        const int row = tileM * 16 + j + half * 8;
        out[(size_t)row * EDIM + colB] = acc[j] + boc;
    }
}

// ---------------------------------------------------------------------------
extern "C" void kernel_launch(void* const* d_in, const int* in_sizes, int n_in,
                              void* d_out, int out_size, void* d_ws, size_t ws_size,
                              hipStream_t stream) {
    const float* x  = (const float*)d_in[0];
    const float* Si = (const float*)d_in[1];
    const float* Zi = (const float*)d_in[2];
    const float* Wq = (const float*)d_in[3];
    const float* bq = (const float*)d_in[4];
    const float* Wk = (const float*)d_in[5];
    const float* bk = (const float*)d_in[6];
    const float* Wv = (const float*)d_in[7];
    const float* bv = (const float*)d_in[8];
    const float* Wo = (const float*)d_in[9];
    const float* bo = (const float*)d_in[10];

    float* out    = (float*)d_out;          // (2048,256)
    float* Si_new = out + OFF_SI;           // (2048,256,256)
    float* Zi_new = out + OFF_ZI;           // (2048,256)

    // workspace: kq | kk | v | V  (4 x 2 MB = 8 MB)
    float* ws    = (float*)d_ws;
    float* ws_kq = ws;
    float* ws_kk = ws + (size_t)RTOT * EDIM;
    float* ws_v  = ws + 2 * (size_t)RTOT * EDIM;
    float* ws_V  = ws + 3 * (size_t)RTOT * EDIM;

    // 2048 output tiles (16x16) per GEMM, 8 waves/block -> 256 blocks
    la_proj_kernel<<<256, 256, 0, stream>>>(x, Zi, Wq, bq, Wk, bk, Wv, bv,
                                            ws_kq, ws_kk, ws_v, Zi_new);
    la_state_kernel<<<RTOT, 256, 0, stream>>>(Si, ws_kq, ws_kk, ws_v, Zi_new,
                                              Si_new, ws_V);
    la_out_kernel<<<256, 256, 0, stream>>>(ws_V, Wo, bo, out);
}